// SRLEmbeddings_51556787421980
// MI455X (gfx1250) — compile-verified
//
#include <hip/hip_runtime.h>
#include <hip/hip_bf16.h>

typedef float v2f __attribute__((ext_vector_type(2)));
typedef float v8f __attribute__((ext_vector_type(8)));

#define BB 16
#define SS 24
#define LL 128
#define DD 768
#define AA 8
#define TT 8

// flat offsets of the 4 concatenated outputs in d_out
#define OFF_AVG  0
#define OFF_PRED (BB*SS*DD)                    // 294912
#define OFF_ARG0 (OFF_PRED + BB*SS*AA*DD)      // 2654208
#define OFF_ARG1 (OFF_ARG0 + BB*SS*AA*DD)      // 5013504

__global__ __launch_bounds__(256, 2)
void srl_fused_pool_wmma(const float* __restrict__ emb,
                         const int*   __restrict__ amask,
                         const int*   __restrict__ sid,
                         const int*   __restrict__ pred_ids,
                         const int*   __restrict__ a0_ids,
                         const int*   __restrict__ a1_ids,
                         float*       __restrict__ out)
{
    __shared__ int   sh_sid[LL];
    __shared__ int   sh_mask[LL];
    __shared__ int   sh_argid[24][TT];
    __shared__ int   sh_cnt[24][TT];
    __shared__ int   sh_sel[32];
    __shared__ float sh_inv[32];
    __shared__ int   sh_rowoff[32];
    __shared__ float sh_M[32][LL + 4];   // +4 pad: conflict-free A-frag reads

    const int bs  = blockIdx.x;          // 0..383  == b*S + s
    const int tid = threadIdx.x;

    // ---- Phase A: stage sentence ids + attention mask ----
    if (tid < LL) {
        sh_sid[tid]  = sid[bs * LL + tid];
        sh_mask[tid] = amask[bs * LL + tid];
    }
    __syncthreads();

    // ---- Phase B: match counts per (row, token); lane 192 counts mask ----
    if (tid < 24 * TT) {
        const int row = tid >> 3;            // 0..23
        const int t   = tid & 7;
        const int* base = (row < 8) ? pred_ids : ((row < 16) ? a0_ids : a1_ids);
        const int a  = row & 7;
        const int id = base[(bs * AA + a) * TT + t];
        sh_argid[row][t] = id;
        int c = 0;
        if (id != 0) {
            #pragma unroll 4
            for (int l = 0; l < LL; ++l) c += (sh_sid[l] == id) ? 1 : 0;
        }
        sh_cnt[row][t] = c;
    } else if (tid == 192) {
        int c = 0;
        #pragma unroll 4
        for (int l = 0; l < LL; ++l) c += sh_mask[l];
        if (c < 1) c = 1;
        sh_inv[0] = 1.0f / (float)c;
        sh_sel[0] = -1;
    }
    __syncthreads();

    // ---- Phase C: pick LAST valid token per row; output row offsets ----
    if (tid < 24) {
        int sel = -1; float inv = 0.0f;
        for (int t = TT - 1; t >= 0; --t) {
            const int id = sh_argid[tid][t];
            const int c  = sh_cnt[tid][t];
            if (id != 0 && c > 0) { sel = id; inv = 1.0f / (float)c; break; }
        }
        sh_sel[tid + 1] = sel;               // -1 => row stays all-zero
        sh_inv[tid + 1] = inv;
    }
    if (tid < 32) {
        const int r = tid;
        int off;
        if (r == 0)        off = OFF_AVG  + bs * DD;
        else if (r <= 8)   off = OFF_PRED + (bs * AA + (r - 1))  * DD;
        else if (r <= 16)  off = OFF_ARG0 + (bs * AA + (r - 9))  * DD;
        else if (r <= 24)  off = OFF_ARG1 + (bs * AA + (r - 17)) * DD;
        else               off = -1;         // padding rows: discard
        sh_rowoff[r] = off;
    }
    __syncthreads();

    // ---- Phase D: build scaled pooling matrix M[32][128] (1/cnt folded in) ----
    for (int idx = tid; idx < 32 * LL; idx += 256) {
        const int row = idx >> 7;
        const int l   = idx & (LL - 1);
        float v;
        if (row == 0)        v = sh_mask[l] ? sh_inv[0] : 0.0f;
        else if (row <= 24)  v = (sh_sid[l] == sh_sel[row]) ? sh_inv[row] : 0.0f;
        else                 v = 0.0f;
        sh_M[row][l] = v;
    }
    __syncthreads();

    // ---- Phase E: OUT[32,768] = M[32,128] @ E[128,768] via v_wmma_f32_16x16x4_f32
    const int wid  = tid >> 5;          // wave 0..7
    const int lane = tid & 31;
    const int lo   = lane & 15;
    const int hi   = lane >> 4;         // 0/1
    const int colBase = wid * 96;       // disjoint 6x16-column slice per wave

    const float* __restrict__ E = emb + (size_t)bs * LL * DD;

    v8f acc[2][6];
    #pragma unroll
    for (int rt = 0; rt < 2; ++rt)
        #pragma unroll
        for (int j = 0; j < 6; ++j)
            acc[rt][j] = (v8f){};

    // unroll 4 => ~48 outstanding NT b32 loads per wave: enough MLP per WGP
    // (~6 KB/wave in flight) to cover HBM latency at the 23.3 TB/s stream rate.
    #pragma unroll 4
    for (int k = 0; k < LL; k += 4) {
        const int ka = k + 2 * hi;      // K pair handled by this half-wave
        v2f a0, a1;
        a0.x = sh_M[lo][ka];        a0.y = sh_M[lo][ka + 1];
        a1.x = sh_M[16 + lo][ka];   a1.y = sh_M[16 + lo][ka + 1];
        #pragma unroll
        for (int j = 0; j < 6; ++j) {
            const int n = colBase + j * 16 + lo;
            v2f b;
            b.x = __builtin_nontemporal_load(&E[(size_t)ka * DD + n]);
            b.y = __builtin_nontemporal_load(&E[(size_t)(ka + 1) * DD + n]);
            acc[0][j] = __builtin_amdgcn_wmma_f32_16x16x4_f32(
                false, a0, false, b, (short)0, acc[0][j], false, false);
            acc[1][j] = __builtin_amdgcn_wmma_f32_16x16x4_f32(
                false, a1, false, b, (short)0, acc[1][j], false, false);
        }
    }

    // ---- Phase F: scatter 16x16 f32 C/D tiles to the 4 outputs ----
    #pragma unroll
    for (int rt = 0; rt < 2; ++rt) {
        #pragma unroll
        for (int v = 0; v < 8; ++v) {
            const int r   = rt * 16 + 8 * hi + v;   // pool row of this element
            const int off = sh_rowoff[r];
            if (off >= 0) {
                #pragma unroll
                for (int j = 0; j < 6; ++j) {
                    const int col = colBase + j * 16 + lo;
                    __builtin_nontemporal_store(acc[rt][j][v], &out[off + col]);
                }
            }
        }
    }
}

extern "C" void kernel_launch(void* const* d_in, const int* in_sizes, int n_in,
                              void* d_out, int out_size, void* d_ws, size_t ws_size,
                              hipStream_t stream) {
    const float* emb   = (const float*)d_in[0];
    const int*   amask = (const int*)  d_in[1];
    const int*   sids  = (const int*)  d_in[2];
    const int*   pids  = (const int*)  d_in[3];
    const int*   a0    = (const int*)  d_in[4];
    const int*   a1    = (const int*)  d_in[5];
    float* out = (float*)d_out;

    dim3 grid(BB * SS);   // one workgroup per (b,s): emb tile read exactly once
    dim3 block(256);      // 8 wave32s
    hipLaunchKernelGGL(srl_fused_pool_wmma, grid, block, 0, stream,
                       emb, amask, sids, pids, a0, a1, out);
}